// PolicyGradientLoss_86363202388021
// MI455X (gfx1250) — compile-verified
//
#include <hip/hip_runtime.h>
#include <cmath>

// ---------------------------------------------------------------------------
// PolicyGradientLoss on gfx1250 (MI455X): memory-bound streaming reduction.
//   d[t] = reverse discounted scan (hierarchical, log-step linear-recurrence)
//   loss = mean( (logsumexp(w[t,:]) - w[t,a[t]]) * (d[t]-mean(d)) )
// CDNA5 path: GLOBAL_LOAD_ASYNC_TO_LDS_B128/B64 double-buffered tile
// streaming, synchronized with s_wait_asynccnt (ASYNCcnt) + barriers.
// ---------------------------------------------------------------------------

#define GAMMA_F 0.99f
#define A_DIM   18
#define CHUNK   4096
#define TPB     256
#define SEG     16                                 // CHUNK / TPB
#define TILE_ROWS 256
#define TILES_PER_CHUNK (CHUNK / TILE_ROWS)        // 16
#define TILE_FLOATS     (TILE_ROWS * A_DIM)        // 4608 floats = 18432 B
#define B64_UNITS       (TILE_FLOATS / 2)          // 2304
#define B128_UNITS      (TILE_FLOATS / 4)          // 1152

#if defined(__has_builtin)
#if __has_builtin(__builtin_amdgcn_global_load_async_to_lds_b64) && \
    __has_builtin(__builtin_amdgcn_s_wait_asynccnt)
#define USE_ASYNC_LDS 1
#endif
#endif
#ifndef USE_ASYNC_LDS
#define USE_ASYNC_LDS 0
#endif

#if USE_ASYNC_LDS && __has_builtin(__builtin_amdgcn_global_load_async_to_lds_b128)
#define USE_B128 1
#else
#define USE_B128 0
#endif

#if USE_ASYNC_LDS
typedef int vi2 __attribute__((ext_vector_type(2)));
typedef __attribute__((address_space(1))) vi2* gptr_vi2;
typedef __attribute__((address_space(3))) vi2* lptr_vi2;
#endif
#if USE_B128
typedef int vi4 __attribute__((ext_vector_type(4)));
typedef __attribute__((address_space(1))) vi4* gptr_vi4;
typedef __attribute__((address_space(3))) vi4* lptr_vi4;
#endif

template <int N>
__device__ __forceinline__ void wait_async() {
#if USE_ASYNC_LDS
  __builtin_amdgcn_s_wait_asynccnt(N);
#endif
}

#if USE_B128
#define STAGE_COPIES 5
__device__ __forceinline__ void stage_tile(const float* __restrict__ gsrc,
                                           float* __restrict__ ldst, int tid) {
#pragma unroll
  for (int c = 0; c < STAGE_COPIES; ++c) {
    int u = c * TPB + tid;                 // 16-byte unit index
    if (u >= B128_UNITS) u -= 128;         // benign duplicate: keeps count uniform
    __builtin_amdgcn_global_load_async_to_lds_b128(
        (gptr_vi4)(gsrc + (size_t)u * 4), (lptr_vi4)(ldst + (size_t)u * 4), 0, 0);
  }
}
#else
#define STAGE_COPIES 9
__device__ __forceinline__ void stage_tile(const float* __restrict__ gsrc,
                                           float* __restrict__ ldst, int tid) {
#pragma unroll
  for (int c = 0; c < STAGE_COPIES; ++c) {
    const int u = c * TPB + tid;           // 8-byte unit index
#if USE_ASYNC_LDS
    __builtin_amdgcn_global_load_async_to_lds_b64(
        (gptr_vi2)(gsrc + (size_t)u * 2), (lptr_vi2)(ldst + (size_t)u * 2), 0, 0);
#else
    float2 v = *reinterpret_cast<const float2*>(gsrc + (size_t)u * 2);
    *reinterpret_cast<float2*>(ldst + (size_t)u * 2) = v;
#endif
  }
}
#endif

// Log-step inclusive suffix scan of the linear recurrence c = a + p*c_right.
// In:  a = per-thread segment value, pw = per-thread segment scale.
// Out (via sA/sP): sA[t] = sum_{u>=t} (prod scales) a[u], sP[t] = total scale.
__device__ __forceinline__ void suffix_scan(float& a, float& pw,
                                            float* sA, float* sP, int tid) {
  sA[tid] = a; sP[tid] = pw;
  __syncthreads();
#pragma unroll
  for (int d = 1; d < TPB; d <<= 1) {
    float ar = 0.f, pr = 1.f;
    if (tid + d < TPB) { ar = sA[tid + d]; pr = sP[tid + d]; }
    __syncthreads();
    a += pw * ar; pw *= pr;
    sA[tid] = a; sP[tid] = pw;
    __syncthreads();
  }
}

// --- K1: per-chunk discounted total via weighted tree reduce ----------------
__global__ void __launch_bounds__(TPB)
k_chunk_totals(const float* __restrict__ r, float* __restrict__ totals, float g16) {
  __shared__ float sW[TPB];
  const int b = blockIdx.x, tid = threadIdx.x;
  const float* rr = r + (size_t)b * CHUNK + (size_t)tid * SEG;
  float w = 0.f, g = 1.f;
#pragma unroll
  for (int j = 0; j < SEG; ++j) { w += g * rr[j]; g *= GAMMA_F; }
  sW[tid] = w;
  __syncthreads();
  float pw = g16;
#pragma unroll
  for (int d = 1; d < TPB; d <<= 1) {
    if ((tid & (2 * d - 1)) == 0) sW[tid] += pw * sW[tid + d];
    pw *= pw;
    __syncthreads();
  }
  if (tid == 0) totals[b] = sW[0];
}

// --- K2: reverse scan over chunk totals (parallel, one block) ---------------
__global__ void __launch_bounds__(TPB)
k_chunk_scan(const float* __restrict__ totals, float* __restrict__ carry,
             int nchunk, float gC) {
  __shared__ float sA[TPB];
  __shared__ float sP[TPB];
  const int tid = threadIdx.x;          // thread t owns chunks {2t, 2t+1}
  const float t0 = (2 * tid     < nchunk) ? totals[2 * tid]     : 0.f;
  const float t1 = (2 * tid + 1 < nchunk) ? totals[2 * tid + 1] : 0.f;
  float a = t0 + gC * t1;
  float pw = gC * gC;
  suffix_scan(a, pw, sA, sP, tid);
  const float E = (tid + 1 < TPB) ? sA[tid + 1] : 0.f;   // suffix from seg t+1
  if (2 * tid + 1 < nchunk) carry[2 * tid + 1] = E;
  if (2 * tid < nchunk)     carry[2 * tid]     = t1 + gC * E;
}

// --- K3: per-chunk sum of d (for the mean), without storing d ---------------
__global__ void __launch_bounds__(TPB)
k_chunk_dsum(const float* __restrict__ r, const float* __restrict__ carryIn,
             float* __restrict__ dsum, float g16) {
  __shared__ float sA[TPB];
  __shared__ float sP[TPB];
  const int b = blockIdx.x, tid = threadIdx.x;
  const float* rr = r + (size_t)b * CHUNK + (size_t)tid * SEG;
  const float carryChunk = carryIn[b];
  float w = 0.f, g = 1.f;
#pragma unroll
  for (int j = 0; j < SEG; ++j) { w += g * rr[j]; g *= GAMMA_F; }
  float a = w, pw = g16;
  suffix_scan(a, pw, sA, sP, tid);
  float c = (tid + 1 < TPB) ? (sA[tid + 1] + sP[tid + 1] * carryChunk) : carryChunk;
  float s = 0.f;
#pragma unroll
  for (int j = SEG - 1; j >= 0; --j) { c = GAMMA_F * c + rr[j]; s += c; }
  __syncthreads();
  sA[tid] = s;
  __syncthreads();
  for (int st = TPB / 2; st > 0; st >>= 1) {
    if (tid < st) sA[tid] += sA[tid + st];
    __syncthreads();
  }
  if (tid == 0) dsum[b] = sA[0];
}

// --- K4: mean(d) ------------------------------------------------------------
__global__ void __launch_bounds__(TPB)
k_mean(const float* __restrict__ dsum, int nchunk, float invT,
       float* __restrict__ meanOut) {
  __shared__ float sW[TPB];
  const int tid = threadIdx.x;
  float s = 0.f;
  for (int i = tid; i < nchunk; i += TPB) s += dsum[i];
  sW[tid] = s;
  __syncthreads();
  for (int st = TPB / 2; st > 0; st >>= 1) {
    if (tid < st) sW[tid] += sW[tid + st];
    __syncthreads();
  }
  if (tid == 0) meanOut[0] = sW[0] * invT;
}

// --- K5: main fused kernel: recompute d into LDS, stream weight tiles -------
__global__ void __launch_bounds__(TPB)
k_loss(const float* __restrict__ w, const float* __restrict__ r,
       const int* __restrict__ ep_as, const float* __restrict__ carryIn,
       const float* __restrict__ meanPtr, float* __restrict__ partials,
       float g16) {
  __shared__ float sBuf[2][TILE_FLOATS];   // 2 * 18432 B
  __shared__ float sD[CHUNK];              // 16384 B
  __shared__ float sA[TPB];
  __shared__ float sP[TPB];

  const int b = blockIdx.x, tid = threadIdx.x;
  const size_t base = (size_t)b * CHUNK;
  const float* wchunk = w + base * (size_t)A_DIM;

  // Kick off tile-0 HBM->LDS streaming immediately (hides phase-A latency).
  stage_tile(wchunk, &sBuf[0][0], tid);

  // ---- Phase A: per-chunk discounted returns into sD (log-step scan) ----
  const float* rr = r + base + (size_t)tid * SEG;
  const float carryChunk = carryIn[b];
  float wsum = 0.f, g = 1.f;
#pragma unroll
  for (int j = 0; j < SEG; ++j) { wsum += g * rr[j]; g *= GAMMA_F; }
  float a = wsum, pw = g16;
  suffix_scan(a, pw, sA, sP, tid);
  {
    float c = (tid + 1 < TPB) ? (sA[tid + 1] + sP[tid + 1] * carryChunk) : carryChunk;
#pragma unroll
    for (int j = SEG - 1; j >= 0; --j) { c = GAMMA_F * c + rr[j]; sD[tid * SEG + j] = c; }
  }
  const float mean = meanPtr[0];
  __syncthreads();

  // ---- Phase B: double-buffered async tile streaming + NLL ----
  float acc = 0.f;
  for (int t = 0; t < TILES_PER_CHUNK; ++t) {
    const int cur = t & 1;
    if (t + 1 < TILES_PER_CHUNK) {
      stage_tile(wchunk + (size_t)(t + 1) * TILE_FLOATS, &sBuf[cur ^ 1][0], tid);
      wait_async<STAGE_COPIES>();  // in-order completion => tile t is resident
    } else {
      wait_async<0>();
    }
    __syncthreads();

    const int row = t * TILE_ROWS + tid;
    const float2* wr = reinterpret_cast<const float2*>(&sBuf[cur][tid * A_DIM]);
    const int aidx = ep_as[base + row];

    float2 v[A_DIM / 2];
#pragma unroll
    for (int p = 0; p < A_DIM / 2; ++p) v[p] = wr[p];   // single LDS pass

    float m = -INFINITY, wa = 0.f;
#pragma unroll
    for (int p = 0; p < A_DIM / 2; ++p) {
      m = fmaxf(m, fmaxf(v[p].x, v[p].y));
      if (2 * p == aidx) wa = v[p].x;
      if (2 * p + 1 == aidx) wa = v[p].y;
    }
    float se = 0.f;
#pragma unroll
    for (int p = 0; p < A_DIM / 2; ++p)
      se += __expf(v[p].x - m) + __expf(v[p].y - m);

    const float lse = m + __logf(se);
    acc += (lse - wa) * (sD[row] - mean);
    __syncthreads();  // before buffer `cur` is overwritten next iteration
  }

  sA[tid] = acc;
  __syncthreads();
  for (int st = TPB / 2; st > 0; st >>= 1) {
    if (tid < st) sA[tid] += sA[tid + st];
    __syncthreads();
  }
  if (tid == 0) partials[b] = sA[0];
}

// --- K6: final reduce -> scalar mean ---------------------------------------
__global__ void __launch_bounds__(TPB)
k_final(const float* __restrict__ partials, int n, float invT,
        float* __restrict__ out) {
  __shared__ float sW[TPB];
  const int tid = threadIdx.x;
  float s = 0.f;
  for (int i = tid; i < n; i += TPB) s += partials[i];
  sW[tid] = s;
  __syncthreads();
  for (int st = TPB / 2; st > 0; st >>= 1) {
    if (tid < st) sW[tid] += sW[tid + st];
    __syncthreads();
  }
  if (tid == 0) out[0] = sW[0] * invT;
}

extern "C" void kernel_launch(void* const* d_in, const int* in_sizes, int n_in,
                              void* d_out, int out_size, void* d_ws, size_t ws_size,
                              hipStream_t stream) {
  const float* w  = (const float*)d_in[0];   // [T, 18] float32
  const float* r  = (const float*)d_in[1];   // [T] float32
  const int*   as = (const int*)d_in[2];     // [T] int (harness delivers int32)
  const int T = in_sizes[1];
  const int nchunk = T / CHUNK;              // 512 for T = 2^21

  float* ws     = (float*)d_ws;
  float* totals = ws;                        // [nchunk]
  float* carry  = ws + nchunk;               // [nchunk]
  float* dsum   = ws + 2 * nchunk;           // [nchunk]
  float* meanP  = ws + 3 * nchunk;           // [1]
  float* parts  = ws + 3 * nchunk + 1;       // [nchunk]

  const float g16  = (float)pow(0.99, 16.0);
  const float gC   = (float)pow(0.99, (double)CHUNK);
  const float invT = 1.0f / (float)T;

  k_chunk_totals<<<nchunk, TPB, 0, stream>>>(r, totals, g16);
  k_chunk_scan<<<1, TPB, 0, stream>>>(totals, carry, nchunk, gC);
  k_chunk_dsum<<<nchunk, TPB, 0, stream>>>(r, carry, dsum, g16);
  k_mean<<<1, TPB, 0, stream>>>(dsum, nchunk, invT, meanP);
  k_loss<<<nchunk, TPB, 0, stream>>>(w, r, as, carry, meanP, parts, g16);
  k_final<<<1, TPB, 0, stream>>>(parts, nchunk, invT, (float*)d_out);
}